// FocalLoss_15461882265978
// MI455X (gfx1250) — compile-verified
//
#include <hip/hip_runtime.h>
#include <math.h>

#define B_ROWS 8192
#define V_COLS 32000
#define THREADS 256

typedef __attribute__((ext_vector_type(2))) float v2f;
typedef __attribute__((ext_vector_type(8))) float v8f;

// ---------------------------------------------------------------------------
// Kernel 1: one row per 256-thread block (8 wave32s). Single pass over the
// row with float4 (global_load_b128) loads, online softmax normalizer:
//   per float4: mv = max4(v); if mv > m { s *= exp(m - mv); m = mv; }
//               s += exp(v.x-m) + exp(v.y-m) + exp(v.z-m) + exp(v.w-m)
// Exactly one v_exp_f32 per element (4 independent per iteration for ILP).
// Then a deterministic LDS tree-merge of (m, s) pairs and the focal loss.
// ---------------------------------------------------------------------------
__global__ __launch_bounds__(THREADS)
void focal_row_kernel(const float* __restrict__ pred,
                      const int* __restrict__ labels,
                      float* __restrict__ row_loss) {
    const int row = blockIdx.x;
    const int t   = threadIdx.x;
    const float* __restrict__ rowp = pred + (size_t)row * V_COLS;

    // Label logit: issue early on thread 0, latency hidden behind the sweep.
    float xlab = 0.0f;
    if (t == 0) {
        const int lab = labels[row];
        xlab = rowp[lab];
    }

    float m = -INFINITY;
    float s = 0.0f;
    const float4* __restrict__ rp4 = (const float4*)rowp;

    #pragma unroll 4
    for (int i = t; i < V_COLS / 4; i += THREADS) {
        const float4 v = rp4[i];
        const float mv = fmaxf(fmaxf(v.x, v.y), fmaxf(v.z, v.w));
        if (mv > m) {                 // rare after warm-up; exp(-inf)=0 handles init
            s *= __expf(m - mv);
            m = mv;
        }
        s += __expf(v.x - m) + __expf(v.y - m) +
             __expf(v.z - m) + __expf(v.w - m);
    }

    __shared__ float msh[THREADS];
    __shared__ float ssh[THREADS];
    msh[t] = m;
    ssh[t] = s;
    __syncthreads();

    // Deterministic pairwise merge: M = max(m1,m2); S = s1*e^(m1-M) + s2*e^(m2-M)
    for (int off = THREADS / 2; off > 0; off >>= 1) {
        if (t < off) {
            const float m1 = msh[t],       s1 = ssh[t];
            const float m2 = msh[t + off], s2 = ssh[t + off];
            const float M  = fmaxf(m1, m2);
            msh[t] = M;
            ssh[t] = s1 * __expf(m1 - M) + s2 * __expf(m2 - M);
        }
        __syncthreads();
    }

    if (t == 0) {
        const float lse  = msh[0] + __logf(ssh[0]);
        const float logp = xlab - lse;
        const float p    = __expf(logp);
        const float om   = 1.0f - p;
        row_loss[row] = -om * om * logp;   // focal loss, gamma = 2
    }
}

// ---------------------------------------------------------------------------
// Kernel 2: single wave32 (EXEC all-ones, as WMMA requires). Sum the 8192
// per-row losses with V_WMMA_F32_16X16X4_F32:  D += A(16x4 values) x ones(4x16).
// A f32 16x4 layout: lanes 0-15 hold (M=lane, K=0..1), lanes 16-31 hold
// (M=lane-16, K=2..3)  =>  per-lane element offset = (l%16)*4 + (l/16)*2.
// Two independent accumulators break the WMMA->WMMA RAW dependency.
// After the loop D[m][n] = rowsum(m) for all n; column n=0 sits in lanes 0
// (M=0..7 across the 8 acc VGPRs) and 16 (M=8..15). Fully deterministic.
// ---------------------------------------------------------------------------
__global__ __launch_bounds__(32)
void focal_reduce_wmma(const float* __restrict__ row_loss,
                       float* __restrict__ out) {
    const int lane = threadIdx.x;
    const int base = (lane & 15) * 4 + (lane >> 4) * 2;

    v2f ones;
    ones[0] = 1.0f;
    ones[1] = 1.0f;
    v8f c0 = {};
    v8f c1 = {};

    for (int i = 0; i < B_ROWS; i += 128) {
        const float* p0 = row_loss + i + base;
        const float* p1 = row_loss + i + 64 + base;
        v2f a0; a0[0] = p0[0]; a0[1] = p0[1];
        v2f a1; a1[0] = p1[0]; a1[1] = p1[1];
        c0 = __builtin_amdgcn_wmma_f32_16x16x4_f32(
                 /*neg_a=*/false, a0, /*neg_b=*/false, ones,
                 /*c_mod=*/(short)0, c0, /*reuse_a=*/false, /*reuse_b=*/false);
        c1 = __builtin_amdgcn_wmma_f32_16x16x4_f32(
                 false, a1, false, ones, (short)0, c1, false, false);
    }

    const v8f c = c0 + c1;  // elementwise; D[m][n] = rowsum(m), all n equal
    float partial = c[0] + c[1] + c[2] + c[3] + c[4] + c[5] + c[6] + c[7];
    const float total = __shfl(partial, 0) + __shfl(partial, 16);
    if (lane == 0) {
        out[0] = total * (1.0f / (float)B_ROWS);
    }
}

// ---------------------------------------------------------------------------
extern "C" void kernel_launch(void* const* d_in, const int* in_sizes, int n_in,
                              void* d_out, int out_size, void* d_ws, size_t ws_size,
                              hipStream_t stream) {
    (void)in_sizes; (void)n_in; (void)out_size; (void)ws_size;
    const float* pred   = (const float*)d_in[0];
    const int*   labels = (const int*)d_in[1];
    float*       out    = (float*)d_out;
    float*       ws     = (float*)d_ws;   // 8192 floats = 32 KB of scratch

    focal_row_kernel<<<dim3(B_ROWS), dim3(THREADS), 0, stream>>>(pred, labels, ws);
    focal_reduce_wmma<<<dim3(1), dim3(32), 0, stream>>>(ws, out);
}